// animal_ARMA_5334349381774
// MI455X (gfx1250) — compile-verified
//
#include <hip/hip_runtime.h>
#include <cstdint>
#include <cstddef>

#define TRIALS  4096
#define STEPS   8192
#define NSTEP   (STEPS - 3)                    // 8189 noise steps
#define CLEN    64                             // steps per chunk
#define NCHUNK  128                            // ceil(NSTEP/CLEN)
#define LASTLEN (NSTEP - (NCHUNK - 1) * CLEN)  // 61
#define PITCH   (CLEN * 2 + 1)                 // padded LDS row pitch (dwords) = 129
#define LPITCH  (LASTLEN * 2)                  // last-chunk row pitch = 122

typedef unsigned int uint32x4 __attribute__((ext_vector_type(4)));
typedef int          int32x8  __attribute__((ext_vector_type(8)));
typedef int          int32x4  __attribute__((ext_vector_type(4)));

#if defined(__HIP_DEVICE_COMPILE__) && __has_builtin(__builtin_amdgcn_tensor_load_to_lds) && __has_builtin(__builtin_amdgcn_s_wait_tensorcnt)
#define USE_TDM 1
#else
#define USE_TDM 0
#endif

// ---------------------------------------------------------------------------
// TDM 2D tile load: 32 rows (trials) x x_elems f32 each, row stride in elems.
// Padded mode inserts 1 dword after every 128 stored dwords -> LDS pitch 129,
// which makes the per-lane (per-trial) LDS row reads bank-conflict-free.
// ---------------------------------------------------------------------------
__device__ __forceinline__ void load_tile_to_lds(float* lds_tile, const float* gbase,
                                                 int lane, int pitch_dw, int x_elems,
                                                 long long row_stride_elems, bool padded)
{
#if USE_TDM
  uint64_t gaddr  = (uint64_t)(uintptr_t)gbase;
  uint32_t ldsoff = (uint32_t)(uintptr_t)lds_tile;   // low 32 bits of flat-LDS = LDS offset
  uint32x4 g0;
  g0.x = 1u;                                         // count=1 (valid user descriptor)
  g0.y = ldsoff;                                     // lds_addr
  g0.z = (uint32_t)gaddr;                            // global_addr[31:0]
  g0.w = (uint32_t)((gaddr >> 32) & 0x1FFFFFFull) | (2u << 30);  // addr[56:32] | type=2
  uint32_t flags = (2u << 16);                       // data_size = 4 bytes
  if (padded)
    flags |= (1u << 20)                              // pad_enable
           | (6u << 22)                              // pad_interval: 128 dwords
           | (0u << 25);                             // pad_amount: 1 dword
  int32x8 g1;
  g1[0] = (int)flags;                                // wg_mask=0 (not in cluster)
  g1[1] = (int)((uint32_t)x_elems << 16);            // tensor_dim0[15:0] (atomic_addr=0)
  g1[2] = (int)(((uint32_t)x_elems >> 16) | (32u << 16)); // tensor_dim0 hi | tensor_dim1=32
  g1[3] = (int)((uint32_t)x_elems << 16);            // tensor_dim1 hi(0) | tile_dim0
  g1[4] = 32;                                        // tile_dim1=32 | tile_dim2=0
  g1[5] = (int)(uint32_t)row_stride_elems;           // tensor_dim0_stride[31:0]
  g1[6] = (int)(uint32_t)(((uint64_t)row_stride_elems >> 32) & 0xFFFFu); // stride hi | stride1 lo=0
  g1[7] = 0;
  int32x4 gz = {0, 0, 0, 0};
#if defined(__clang_major__) && (__clang_major__ >= 23)
  int32x8 z8 = {0, 0, 0, 0, 0, 0, 0, 0};
  __builtin_amdgcn_tensor_load_to_lds(g0, g1, gz, gz, z8, 0);
#else
  __builtin_amdgcn_tensor_load_to_lds(g0, g1, gz, gz, 0);
#endif
  __builtin_amdgcn_s_wait_tensorcnt(0);
  (void)lane; (void)pitch_dw;
#else
  for (int i = 0; i < x_elems; ++i)
    lds_tile[lane * pitch_dw + i] = gbase[(long long)lane * row_stride_elems + i];
#endif
  __asm__ volatile("" ::: "memory");   // LDS now holds DMA'd data; block reordering
}

// ---------------------------------------------------------------------------
// Phase A: per (trial-group, full chunk) wave, compute particular-solution
// window P (zero initial state) over the chunk. Chunks 0..126 only.
// P layout is SoA: P[(chunk*6 + k)*TRIALS + trial]  -> coalesced everywhere.
// ---------------------------------------------------------------------------
__global__ __launch_bounds__(32) void kPartials(const float* __restrict__ alpha,
                                                const float* __restrict__ sigma,
                                                const float* __restrict__ rho,
                                                const float* __restrict__ mu,
                                                const float* __restrict__ normals,
                                                float* __restrict__ P)
{
  __shared__ float tile[32 * PITCH];
  const int tileId = blockIdx.x;
  const int chunk  = tileId % (NCHUNK - 1);
  const int tg     = tileId / (NCHUNK - 1);
  const int lane   = threadIdx.x;
  const int trial  = tg * 32 + lane;

  const float s0 = sigma[0], s1 = sigma[1], r = rho[0];
  const float l00 = s0, l10 = r * s1, l11 = sqrtf(1.0f - r * r) * s1;
  const float m0 = mu[0], m1 = mu[1];
  const float a00 = alpha[0], a01 = alpha[1];
  const float a10 = alpha[2], a11 = alpha[3];
  const float a20 = alpha[4], a21 = alpha[5];

  const float* gbase = normals + (long long)tg * 32 * (NSTEP * 2) + (long long)chunk * (CLEN * 2);
  load_tile_to_lds(tile, gbase, lane, PITCH, CLEN * 2, (long long)NSTEP * 2, true);

  const float* row = &tile[lane * PITCH];
  float w00 = 0.f, w01 = 0.f, w02 = 0.f, w10 = 0.f, w11 = 0.f, w12 = 0.f;
#pragma unroll 8
  for (int j = 0; j < CLEN; ++j) {
    float z0 = row[2 * j], z1 = row[2 * j + 1];
    float e0 = fmaf(l00, z0, m0);
    float e1 = fmaf(l10, z0, fmaf(l11, z1, m1));
    float n0 = fmaf(a00, w00, fmaf(a10, w01, fmaf(a20, w02, e0)));
    float n1 = fmaf(a01, w10, fmaf(a11, w11, fmaf(a21, w12, e1)));
    w00 = w01; w01 = w02; w02 = n0;
    w10 = w11; w11 = w12; w12 = n1;
  }
  float* p = &P[(size_t)chunk * 6 * TRIALS + trial];
  p[0 * TRIALS] = w00; p[1 * TRIALS] = w01; p[2 * TRIALS] = w02;
  p[3 * TRIALS] = w10; p[4 * TRIALS] = w11; p[5 * TRIALS] = w12;
}

// ---------------------------------------------------------------------------
// Phase B: per trial, chain chunk-start windows: w <- M^64 w + P[c].
// M^64 computed in-registers by 6 squarings of the 3x3 companion matrix.
// Also writes the first ORDER outputs (= x_0 exactly).
// ---------------------------------------------------------------------------
__device__ __forceinline__ void sq3(float* M)
{
  float T[9];
#pragma unroll
  for (int i = 0; i < 3; ++i)
#pragma unroll
    for (int j = 0; j < 3; ++j)
      T[i * 3 + j] = M[i * 3 + 0] * M[0 * 3 + j] + M[i * 3 + 1] * M[1 * 3 + j] + M[i * 3 + 2] * M[2 * 3 + j];
#pragma unroll
  for (int k = 0; k < 9; ++k) M[k] = T[k];
}

__global__ __launch_bounds__(256) void kPrefix(const float* __restrict__ alpha,
                                               const float* __restrict__ xmu,
                                               const float* __restrict__ x0,
                                               const float* __restrict__ P,
                                               float* __restrict__ S,
                                               float* __restrict__ out)
{
  const int trial = blockIdx.x * 256 + threadIdx.x;
  const float xm0 = xmu[0], xm1 = xmu[1];

  const float* xr = &x0[(size_t)trial * 6];
  float w00 = xr[0] - xm0, w10 = xr[1] - xm1;
  float w01 = xr[2] - xm0, w11 = xr[3] - xm1;
  float w02 = xr[4] - xm0, w12 = xr[5] - xm1;

  // first ORDER outputs: dx + xmu == x_0
  float* o = &out[(size_t)trial * STEPS * 2];
#pragma unroll
  for (int k = 0; k < 6; ++k) o[k] = xr[k];

  float M0[9] = {0.f, 1.f, 0.f, 0.f, 0.f, 1.f, alpha[0], alpha[2], alpha[4]};
  float M1[9] = {0.f, 1.f, 0.f, 0.f, 0.f, 1.f, alpha[1], alpha[3], alpha[5]};
#pragma unroll
  for (int s = 0; s < 6; ++s) { sq3(M0); sq3(M1); }   // companion^64

  for (int c = 0; c < NCHUNK; ++c) {
    float* sv = &S[(size_t)c * 6 * TRIALS + trial];
    sv[0 * TRIALS] = w00; sv[1 * TRIALS] = w01; sv[2 * TRIALS] = w02;
    sv[3 * TRIALS] = w10; sv[4 * TRIALS] = w11; sv[5 * TRIALS] = w12;
    if (c < NCHUNK - 1) {
      const float* p = &P[(size_t)c * 6 * TRIALS + trial];
      float p0 = p[0 * TRIALS], p1 = p[1 * TRIALS], p2 = p[2 * TRIALS];
      float p3 = p[3 * TRIALS], p4 = p[4 * TRIALS], p5 = p[5 * TRIALS];
      float n00 = M0[0] * w00 + M0[1] * w01 + M0[2] * w02 + p0;
      float n01 = M0[3] * w00 + M0[4] * w01 + M0[5] * w02 + p1;
      float n02 = M0[6] * w00 + M0[7] * w01 + M0[8] * w02 + p2;
      float n10 = M1[0] * w10 + M1[1] * w11 + M1[2] * w12 + p3;
      float n11 = M1[3] * w10 + M1[4] * w11 + M1[5] * w12 + p4;
      float n12 = M1[6] * w10 + M1[7] * w11 + M1[8] * w12 + p5;
      w00 = n00; w01 = n01; w02 = n02; w10 = n10; w11 = n11; w12 = n12;
    }
  }
}

// ---------------------------------------------------------------------------
// Phase C: per (trial-group, chunk) wave: TDM tile load, run recurrence from
// the chunk-start window S, stage outputs in place in LDS, cooperative
// coalesced writeout. Tail chunk (61 steps) uses exact-size unpadded tile.
// ---------------------------------------------------------------------------
__global__ __launch_bounds__(32) void kEmit(const float* __restrict__ alpha,
                                            const float* __restrict__ xmu,
                                            const float* __restrict__ sigma,
                                            const float* __restrict__ rho,
                                            const float* __restrict__ mu,
                                            const float* __restrict__ normals,
                                            const float* __restrict__ S,
                                            float* __restrict__ out)
{
  __shared__ float tile[32 * PITCH];
  const int tileId = blockIdx.x;
  const int chunk  = tileId % NCHUNK;
  const int tg     = tileId / NCHUNK;
  const int lane   = threadIdx.x;
  const int trial  = tg * 32 + lane;
  const bool full  = (chunk != NCHUNK - 1);

  const float s0 = sigma[0], s1 = sigma[1], r = rho[0];
  const float l00 = s0, l10 = r * s1, l11 = sqrtf(1.0f - r * r) * s1;
  const float m0 = mu[0], m1 = mu[1];
  const float a00 = alpha[0], a01 = alpha[1];
  const float a10 = alpha[2], a11 = alpha[3];
  const float a20 = alpha[4], a21 = alpha[5];
  const float xm0 = xmu[0], xm1 = xmu[1];

  const float* gbase = normals + (long long)tg * 32 * (NSTEP * 2) + (long long)chunk * (CLEN * 2);
  const float* sv = &S[(size_t)chunk * 6 * TRIALS + trial];
  float w00 = sv[0 * TRIALS], w01 = sv[1 * TRIALS], w02 = sv[2 * TRIALS];
  float w10 = sv[3 * TRIALS], w11 = sv[4 * TRIALS], w12 = sv[5 * TRIALS];

  if (full) {
    load_tile_to_lds(tile, gbase, lane, PITCH, CLEN * 2, (long long)NSTEP * 2, true);
    float* row = &tile[lane * PITCH];
#pragma unroll 8
    for (int j = 0; j < CLEN; ++j) {
      float z0 = row[2 * j], z1 = row[2 * j + 1];
      float e0 = fmaf(l00, z0, m0);
      float e1 = fmaf(l10, z0, fmaf(l11, z1, m1));
      float n0 = fmaf(a00, w00, fmaf(a10, w01, fmaf(a20, w02, e0)));
      float n1 = fmaf(a01, w10, fmaf(a11, w11, fmaf(a21, w12, e1)));
      w00 = w01; w01 = w02; w02 = n0;
      w10 = w11; w11 = w12; w12 = n1;
      row[2 * j]     = n0 + xm0;   // stage output in place over consumed noise
      row[2 * j + 1] = n1 + xm1;
    }
    __asm__ volatile("" ::: "memory");   // per-lane LDS writes before cross-lane reads
    // cooperative, lane-contiguous global writeout: one trial row (512B) per step
#pragma unroll 4
    for (int rr = 0; rr < 32; ++rr) {
      float v0 = tile[rr * PITCH + lane * 4 + 0];
      float v1 = tile[rr * PITCH + lane * 4 + 1];
      float v2 = tile[rr * PITCH + lane * 4 + 2];
      float v3 = tile[rr * PITCH + lane * 4 + 3];
      size_t obase = ((size_t)(tg * 32 + rr) * STEPS + 3 + (size_t)chunk * CLEN) * 2 + lane * 4;
      *reinterpret_cast<float2*>(&out[obase])     = make_float2(v0, v1);
      *reinterpret_cast<float2*>(&out[obase + 2]) = make_float2(v2, v3);
    }
  } else {
    load_tile_to_lds(tile, gbase, lane, LPITCH, LASTLEN * 2, (long long)NSTEP * 2, false);
    const float* row = &tile[lane * LPITCH];
    float* orow = &out[((size_t)trial * STEPS + 3 + (size_t)chunk * CLEN) * 2];
#pragma unroll 4
    for (int j = 0; j < LASTLEN; ++j) {
      float z0 = row[2 * j], z1 = row[2 * j + 1];
      float e0 = fmaf(l00, z0, m0);
      float e1 = fmaf(l10, z0, fmaf(l11, z1, m1));
      float n0 = fmaf(a00, w00, fmaf(a10, w01, fmaf(a20, w02, e0)));
      float n1 = fmaf(a01, w10, fmaf(a11, w11, fmaf(a21, w12, e1)));
      w00 = w01; w01 = w02; w02 = n0;
      w10 = w11; w11 = w12; w12 = n1;
      *reinterpret_cast<float2*>(&orow[2 * j]) = make_float2(n0 + xm0, n1 + xm1);
    }
  }
}

// ---------------------------------------------------------------------------
// Fallback: plain single-pass scan, thread per trial (used only if d_ws is
// too small for the blocked-scan scratch).
// ---------------------------------------------------------------------------
__global__ __launch_bounds__(64) void kSimple(const float* __restrict__ alpha,
                                              const float* __restrict__ xmu,
                                              const float* __restrict__ sigma,
                                              const float* __restrict__ rho,
                                              const float* __restrict__ mu,
                                              const float* __restrict__ x0,
                                              const float* __restrict__ normals,
                                              float* __restrict__ out)
{
  const int trial = blockIdx.x * 64 + threadIdx.x;
  const float s0 = sigma[0], s1 = sigma[1], r = rho[0];
  const float l00 = s0, l10 = r * s1, l11 = sqrtf(1.0f - r * r) * s1;
  const float m0 = mu[0], m1 = mu[1];
  const float a00 = alpha[0], a01 = alpha[1];
  const float a10 = alpha[2], a11 = alpha[3];
  const float a20 = alpha[4], a21 = alpha[5];
  const float xm0 = xmu[0], xm1 = xmu[1];

  const float* xr = &x0[(size_t)trial * 6];
  float* o = &out[(size_t)trial * STEPS * 2];
#pragma unroll
  for (int k = 0; k < 6; ++k) o[k] = xr[k];

  float w00 = xr[0] - xm0, w10 = xr[1] - xm1;
  float w01 = xr[2] - xm0, w11 = xr[3] - xm1;
  float w02 = xr[4] - xm0, w12 = xr[5] - xm1;

  const float* nr = normals + (size_t)trial * NSTEP * 2;
  float* orow = o + 6;
  for (int n = 0; n < NSTEP; ++n) {
    float2 z = *reinterpret_cast<const float2*>(&nr[2 * n]);
    float e0 = fmaf(l00, z.x, m0);
    float e1 = fmaf(l10, z.x, fmaf(l11, z.y, m1));
    float n0 = fmaf(a00, w00, fmaf(a10, w01, fmaf(a20, w02, e0)));
    float n1 = fmaf(a01, w10, fmaf(a11, w11, fmaf(a21, w12, e1)));
    w00 = w01; w01 = w02; w02 = n0;
    w10 = w11; w11 = w12; w12 = n1;
    *reinterpret_cast<float2*>(&orow[2 * n]) = make_float2(n0 + xm0, n1 + xm1);
  }
}

// ---------------------------------------------------------------------------
extern "C" void kernel_launch(void* const* d_in, const int* in_sizes, int n_in,
                              void* d_out, int out_size, void* d_ws, size_t ws_size,
                              hipStream_t stream)
{
  const float* alpha   = (const float*)d_in[0];  // (3,2)
  const float* xmu     = (const float*)d_in[1];  // (2,)
  const float* sigma   = (const float*)d_in[2];  // (2,)
  const float* rho     = (const float*)d_in[3];  // scalar
  const float* mu      = (const float*)d_in[4];  // (2,)
  const float* x0      = (const float*)d_in[5];  // (4096,3,2)
  const float* normals = (const float*)d_in[6];  // (4096,8189,2)
  float* out = (float*)d_out;                    // (4096,8192,2)

  const size_t pFloats = (size_t)(NCHUNK - 1) * TRIALS * 6;
  const size_t sFloats = (size_t)NCHUNK * TRIALS * 6;

  if (ws_size >= (pFloats + sFloats) * sizeof(float)) {
    float* P = (float*)d_ws;
    float* S = P + pFloats;
    kPartials<<<128 * (NCHUNK - 1), 32, 0, stream>>>(alpha, sigma, rho, mu, normals, P);
    kPrefix<<<TRIALS / 256, 256, 0, stream>>>(alpha, xmu, x0, P, S, out);
    kEmit<<<128 * NCHUNK, 32, 0, stream>>>(alpha, xmu, sigma, rho, mu, normals, S, out);
  } else {
    kSimple<<<TRIALS / 64, 64, 0, stream>>>(alpha, xmu, sigma, rho, mu, x0, normals, out);
  }
}